// DIG_59468117181018
// MI455X (gfx1250) — compile-verified
//
#include <hip/hip_runtime.h>
#include <stdint.h>

#define TPB 256
#define TILE 256   // edges staged per block-iteration (1 edge per thread)

// ---------------------------------------------------------------------------
// CDNA5 primitives (inline asm; see CDNA5 ISA, CDNA5 ISA)
// ---------------------------------------------------------------------------

// Fire-and-forget device-scope f32 atomic add (no return -> STOREcnt, never
// stalls the issuing wave).
__device__ __forceinline__ void atomAddF32(float* p, float v) {
  asm volatile("global_atomic_add_f32 %0, %1, off scope:SCOPE_DEV"
               :: "v"(p), "v"(v)
               : "memory");
}

// Async global -> LDS copy, 4 bytes per lane (GV mode, 64-bit vaddr).
// VDST = per-lane LDS byte offset, VADDR = per-lane 64-bit global address.
__device__ __forceinline__ void asyncG2L_b32(void* lds, const void* g) {
  uint32_t l = (uint32_t)(uintptr_t)lds;   // low 32 bits of flat ptr == LDS offset
  asm volatile("global_load_async_to_lds_b32 %0, %1, off"
               :: "v"(l), "v"(g)
               : "memory");
}

#define WAIT_ASYNC(n) asm volatile("s_wait_asynccnt " #n ::: "memory")

// ---------------------------------------------------------------------------
// SpMM: y[dst] += norm * x[src]  (one wave per edge, 32 lanes x float2 = 64 dims)
// Edge metadata (src,dst,norm) is double-buffer streamed into LDS with async
// global->LDS copies. Each wave stages and reads ONLY its own 32-edge slice,
// so no block barriers are needed; per-wave s_wait_asynccnt pipelines tiles.
// (Defined first so the disasm snippet shows the hot kernel.)
// ---------------------------------------------------------------------------

__global__ __launch_bounds__(TPB, 2)
void spmm_kernel(const float* __restrict__ x, float* __restrict__ y,
                 const int* __restrict__ srcI, const int* __restrict__ dstI,
                 const float* __restrict__ nrmI, int nE, int ntiles) {
  __shared__ int   sS[2][TILE];
  __shared__ int   sD[2][TILE];
  __shared__ float sW[2][TILE];

  const int tid   = threadIdx.x;
  const int lane  = tid & 31;
  const int wbase = tid & ~31;   // wave_id * 32

  auto stage = [&](int t, int b) {
    int e = t * TILE + tid;
    if (e >= nE) e = nE - 1;     // clamp: stays in-bounds, never consumed
    asyncG2L_b32(&sS[b][tid], srcI + e);
    asyncG2L_b32(&sD[b][tid], dstI + e);
    asyncG2L_b32(&sW[b][tid], nrmI + e);
  };

  int t = (int)blockIdx.x;
  if (t >= ntiles) return;
  stage(t, 0);
  int buf = 0;

  for (;;) {
    const int  tn        = t + (int)gridDim.x;
    const bool have_next = (tn < ntiles);
    if (have_next) {
      stage(tn, buf ^ 1);            // prefetch next tile into the other buffer
      WAIT_ASYNC(0x3);               // 3 outstanding belong to next tile
    } else {
      WAIT_ASYNC(0x0);
    }

    const int ebase = t * TILE + wbase;
    const int cnt   = min(32, nE - ebase);
    #pragma unroll 4
    for (int j = 0; j < cnt; ++j) {
      const int   s = sS[buf][wbase + j];
      const int   d = sD[buf][wbase + j];
      const float w = sW[buf][wbase + j];
      // prefetch the gather row a few edges ahead
      const int jp = j + 4;
      if (jp < cnt) {
        const int sp = sS[buf][wbase + jp];
        __builtin_prefetch(x + ((size_t)sp << 6) + (lane << 1), 0, 0);
      }
      const float2 v = *(((const float2*)(x + ((size_t)s << 6))) + lane);
      float* yp = y + ((size_t)d << 6) + (lane << 1);
      atomAddF32(yp,     w * v.x);
      atomAddF32(yp + 1, w * v.y);
    }

    if (!have_next) break;
    t = tn;
    buf ^= 1;
  }
}

// ---------------------------------------------------------------------------
// Precompute: degree, rsqrt, per-edge norm
// ---------------------------------------------------------------------------

__global__ __launch_bounds__(TPB)
void zero_f32_kernel(float* __restrict__ p, int n) {
  int i = blockIdx.x * TPB + threadIdx.x;
  if (i < n) p[i] = 0.0f;
}

__global__ __launch_bounds__(TPB)
void deg_kernel(const int* __restrict__ dst, float* __restrict__ deg, int nE) {
  int e = blockIdx.x * TPB + threadIdx.x;
  if (e < nE) atomAddF32(&deg[dst[e]], 1.0f);
}

__global__ __launch_bounds__(TPB)
void rsqrt_kernel(float* __restrict__ deg, int n) {
  int i = blockIdx.x * TPB + threadIdx.x;
  if (i < n) {
    float d = deg[i];
    deg[i] = (d > 0.0f) ? rsqrtf(d) : 0.0f;
  }
}

__global__ __launch_bounds__(TPB)
void norm_kernel(const int* __restrict__ src, const int* __restrict__ dst,
                 const float* __restrict__ dis, float* __restrict__ nrm, int nE) {
  int e = blockIdx.x * TPB + threadIdx.x;
  if (e < nE) nrm[e] = dis[src[e]] * dis[dst[e]];
}

// ---------------------------------------------------------------------------
// Per-embedding elementwise kernels (float4 vectorized; D=64 -> 16 float4/row)
// ---------------------------------------------------------------------------

// cur = concat(user,item); out = cur (emb0 term); nxt = 0
__global__ __launch_bounds__(TPB)
void init_kernel(const float* __restrict__ ue, const float* __restrict__ ie,
                 int nUser, int n4,
                 float* __restrict__ cur, float* __restrict__ nxt,
                 float* __restrict__ out) {
  int i = blockIdx.x * TPB + threadIdx.x;
  if (i >= n4) return;
  int row = i >> 4;           // 16 float4 per 64-float row
  int c   = i & 15;
  const float4* sp = (row < nUser)
      ? ((const float4*)ue) + ((size_t)row * 16 + c)
      : ((const float4*)ie) + ((size_t)(row - nUser) * 16 + c);
  float4 v = *sp;
  ((float4*)cur)[i] = v;
  ((float4*)out)[i] = v;
  ((float4*)nxt)[i] = make_float4(0.f, 0.f, 0.f, 0.f);
}

// out += nxt; zero the buffer that becomes next layer's destination
__global__ __launch_bounds__(TPB)
void accum_kernel(float* __restrict__ out, const float* __restrict__ nxt,
                  float* __restrict__ zbuf, int n4) {
  int i = blockIdx.x * TPB + threadIdx.x;
  if (i >= n4) return;
  float4 o = ((float4*)out)[i];
  float4 a = ((const float4*)nxt)[i];
  o.x += a.x; o.y += a.y; o.z += a.z; o.w += a.w;
  ((float4*)out)[i] = o;
  ((float4*)zbuf)[i] = make_float4(0.f, 0.f, 0.f, 0.f);
}

// out = (out + nxt) / (L+1)^2
__global__ __launch_bounds__(TPB)
void final_kernel(float* __restrict__ out, const float* __restrict__ nxt, int n4) {
  int i = blockIdx.x * TPB + threadIdx.x;
  if (i >= n4) return;
  const float s = 1.0f / 16.0f;
  float4 o = ((float4*)out)[i];
  float4 a = ((const float4*)nxt)[i];
  o.x = (o.x + a.x) * s; o.y = (o.y + a.y) * s;
  o.z = (o.z + a.z) * s; o.w = (o.w + a.w) * s;
  ((float4*)out)[i] = o;
}

// ---------------------------------------------------------------------------
// Host orchestration
// ---------------------------------------------------------------------------

static inline int cdiv_i(int a, int b) { return (a + b - 1) / b; }

extern "C" void kernel_launch(void* const* d_in, const int* in_sizes, int n_in,
                              void* d_out, int out_size, void* d_ws, size_t ws_size,
                              hipStream_t stream) {
  (void)n_in; (void)out_size; (void)ws_size;

  const float* user_int = (const float*)d_in[0];
  const float* item_int = (const float*)d_in[1];
  const float* user_geo = (const float*)d_in[2];
  const float* item_geo = (const float*)d_in[3];
  const int*   eidx     = (const int*)d_in[4];

  const int D     = 64;
  const int nUser = in_sizes[0] / D;
  const int nItem = in_sizes[1] / D;
  const int N     = nUser + nItem;
  const int E     = in_sizes[4] / 2;

  const int* src = eidx;        // edge_index[0]
  const int* dst = eidx + E;    // edge_index[1]

  // workspace layout (floats): dis[N] | norm[E] | bufA[N*D] | bufB[N*D]
  float* ws   = (float*)d_ws;
  float* dis  = ws;
  float* nrm  = dis + N;
  float* bufA = nrm + E;
  float* bufB = bufA + (size_t)N * D;

  float* outI = (float*)d_out;             // int embedding result [N*D]
  float* outG = outI + (size_t)N * D;      // geo embedding result [N*D]

  const int n4     = N * (D / 4);          // float4 elements per node table
  const int gN     = cdiv_i(N, TPB);
  const int gE     = cdiv_i(E, TPB);
  const int g4     = cdiv_i(n4, TPB);
  const int ntiles = cdiv_i(E, TILE);
  const int gSpmm  = (ntiles < 2048) ? ntiles : 2048;

  // --- shared precompute: symmetric gcn_norm ---
  zero_f32_kernel<<<gN, TPB, 0, stream>>>(dis, N);
  deg_kernel     <<<gE, TPB, 0, stream>>>(dst, dis, E);
  rsqrt_kernel   <<<gN, TPB, 0, stream>>>(dis, N);
  norm_kernel    <<<gE, TPB, 0, stream>>>(src, dst, dis, nrm, E);

  // --- per embedding: 3-layer LGConv with running accumulator in d_out ---
  for (int emb = 0; emb < 2; ++emb) {
    const float* ue  = emb ? user_geo : user_int;
    const float* ie  = emb ? item_geo : item_int;
    float*       out = emb ? outG     : outI;

    init_kernel <<<g4, TPB, 0, stream>>>(ue, ie, nUser, n4, bufA, bufB, out);

    // layer 1: bufA -> bufB
    spmm_kernel <<<gSpmm, TPB, 0, stream>>>(bufA, bufB, src, dst, nrm, E, ntiles);
    accum_kernel<<<g4, TPB, 0, stream>>>(out, bufB, bufA, n4);   // out += e1; zero bufA
    // layer 2: bufB -> bufA
    spmm_kernel <<<gSpmm, TPB, 0, stream>>>(bufB, bufA, src, dst, nrm, E, ntiles);
    accum_kernel<<<g4, TPB, 0, stream>>>(out, bufA, bufB, n4);   // out += e2; zero bufB
    // layer 3: bufA -> bufB, then scale by 1/16
    spmm_kernel <<<gSpmm, TPB, 0, stream>>>(bufA, bufB, src, dst, nrm, E, ntiles);
    final_kernel<<<g4, TPB, 0, stream>>>(out, bufB, n4);
  }
}